// LightGCN_27711128994136
// MI455X (gfx1250) — compile-verified
//
#include <hip/hip_runtime.h>

#define NUM_USERS 60000
#define NUM_ITEMS 40000
#define N_ENT     (NUM_USERS + NUM_ITEMS)   // 100000
#define D_DIM     64
#define BUCKET    2048
#define Q_PER_ROW 49                        // ceil(100000/2048)
#define N_LAYERS  3

typedef __attribute__((ext_vector_type(2))) float v2f;
typedef __attribute__((ext_vector_type(8))) float v8f;

// ---------------------------------------------------------------------------
// Kernel 1: compositional embedding build.  all_emb[e] = Q[e/49] + R[e&2047]
// Written to BOTH emb0 (layer-0 input) and acc (layer accumulator).
// One thread per (entity, float4-chunk): N_ENT*16 threads.
// ---------------------------------------------------------------------------
__global__ void cerp_build_kernel(const float* __restrict__ Q,
                                  const float* __restrict__ R,
                                  float* __restrict__ emb0,
                                  float* __restrict__ acc) {
    int tid = blockIdx.x * blockDim.x + threadIdx.x;
    int e = tid >> 4;
    int c = tid & 15;
    if (e >= N_ENT) return;
    int q = e / Q_PER_ROW;          // < 2048
    int r = e & (BUCKET - 1);
    float4 a = ((const float4*)(Q + (size_t)q * D_DIM))[c];
    float4 b = ((const float4*)(R + (size_t)r * D_DIM))[c];
    float4 s;
    s.x = a.x + b.x; s.y = a.y + b.y; s.z = a.z + b.z; s.w = a.w + b.w;
    ((float4*)(emb0 + (size_t)e * D_DIM))[c] = s;
    ((float4*)(acc  + (size_t)e * D_DIM))[c] = s;
}

// ---------------------------------------------------------------------------
// Zero a buffer (float4 granularity).
// ---------------------------------------------------------------------------
__global__ void zero_kernel(float4* __restrict__ p, int n4) {
    int tid = blockIdx.x * blockDim.x + threadIdx.x;
    if (tid < n4) { float4 z; z.x = z.y = z.z = z.w = 0.0f; p[tid] = z; }
}

// ---------------------------------------------------------------------------
// Kernel 2: COO SpMM scatter.  dst[row] += src[col] * val   (segment_sum)
// 16 threads per edge, each owning one float4 chunk of the 64-dim row.
// All three buffers are L2-resident (77 MB < 192 MB L2), so the random
// gathers and the no-return global_atomic_add_f32 resolve on-chip.
// ---------------------------------------------------------------------------
__global__ void spmm_scatter_kernel(const float* __restrict__ src,
                                    float* __restrict__ dst,
                                    const float* __restrict__ vals,
                                    const int* __restrict__ rows,
                                    const int* __restrict__ cols,
                                    int nnz) {
    long long tid = (long long)blockIdx.x * blockDim.x + threadIdx.x;
    long long e = tid >> 4;
    int c = (int)(tid & 15);
    if (e >= nnz) return;
    int   row = rows[e];
    int   col = cols[e];
    float v   = vals[e];
    float4 x = ((const float4*)(src + (size_t)col * D_DIM))[c];
    float* d = dst + (size_t)row * D_DIM + c * 4;
    atomicAdd(d + 0, x.x * v);
    atomicAdd(d + 1, x.y * v);
    atomicAdd(d + 2, x.z * v);
    atomicAdd(d + 3, x.w * v);
}

// ---------------------------------------------------------------------------
// Kernel 3: acc += layer output (float4 granularity).
// ---------------------------------------------------------------------------
__global__ void accum_kernel(float4* __restrict__ acc,
                             const float4* __restrict__ d, int n4) {
    int tid = blockIdx.x * blockDim.x + threadIdx.x;
    if (tid < n4) {
        float4 a = acc[tid];
        float4 b = d[tid];
        a.x += b.x; a.y += b.y; a.z += b.z; a.w += b.w;
        acc[tid] = a;
    }
}

// ---------------------------------------------------------------------------
// Kernel 4: gamma[b] = dot(acc[users[b]], acc[NUM_USERS+items[b]]) / 16
// via V_WMMA_F32_16X16X4_F32: each wave computes a 16x16 = U·V^T tile with
// 16 WMMAs over K=64 and extracts the diagonal through LDS.
//   A 16x4 f32 layout: lane m = lane&15, K = k0 + 2*(lane>>4) + {0,1}
//   B 4x16 f32 layout: lane n = lane&15, same K pattern  (B[K][n] = V[n][K])
//   D 16x16 f32:  VGPR r, lanes 0-15 -> M=r,   lanes 16-31 -> M=r+8
// 4 waves per 128-thread block -> 64 batch rows per block.
// ---------------------------------------------------------------------------
__global__ void gamma_wmma_kernel(const float* __restrict__ acc,
                                  const int* __restrict__ users,
                                  const int* __restrict__ items,
                                  float* __restrict__ out) {
    __shared__ float cbuf[4 * 8 * 32];
    int wave = threadIdx.x >> 5;
    int lane = threadIdx.x & 31;
    int base = (blockIdx.x * 4 + wave) * 16;

    int m  = lane & 15;
    int kk = (lane >> 4) << 1;              // 0 for lanes 0-15, 2 for 16-31

    int u  = users[base + m];
    int it = items[base + m];
    const float* urow = acc + (size_t)u * D_DIM;
    const float* irow = acc + ((size_t)NUM_USERS + (size_t)it) * D_DIM;

    v8f c = {};
#pragma unroll
    for (int k0 = 0; k0 < D_DIM; k0 += 4) {
        v2f a, b;
        a.x = urow[k0 + kk]; a.y = urow[k0 + kk + 1];
        b.x = irow[k0 + kk]; b.y = irow[k0 + kk + 1];
        c = __builtin_amdgcn_wmma_f32_16x16x4_f32(
                /*neg_a=*/false, a, /*neg_b=*/false, b,
                /*c_mod=*/(short)0, c, /*reuse_a=*/false, /*reuse_b=*/false);
    }

    // Spill the 16x16 accumulator tile to LDS, pull out the diagonal.
    float* buf = cbuf + wave * 256;
#pragma unroll
    for (int r = 0; r < 8; ++r) buf[r * 32 + lane] = c[r];
    __syncthreads();

    if (lane < 16) {
        int i = lane;
        float d = (i < 8) ? buf[i * 32 + i]                 // M=i (VGPR i), N=i
                          : buf[(i - 8) * 32 + (i + 16)];   // M=i (VGPR i-8), N=i
        // light_out = acc/4 on both operands -> gamma = dot/16
        out[base + i] = d * (1.0f / 16.0f);
    }
}

// ---------------------------------------------------------------------------
extern "C" void kernel_launch(void* const* d_in, const int* in_sizes, int n_in,
                              void* d_out, int out_size, void* d_ws, size_t ws_size,
                              hipStream_t stream) {
    (void)n_in; (void)out_size; (void)ws_size;

    const float* Q     = (const float*)d_in[0];
    const float* R     = (const float*)d_in[1];
    const float* vals  = (const float*)d_in[2];
    const int*   rows  = (const int*)d_in[3];
    const int*   cols  = (const int*)d_in[4];
    const int*   users = (const int*)d_in[5];
    const int*   items = (const int*)d_in[6];
    float*       out   = (float*)d_out;

    int nnz   = in_sizes[2];
    int batch = in_sizes[5];

    size_t embElems = (size_t)N_ENT * D_DIM;     // 6.4M floats = 25.6 MB
    float* E0  = (float*)d_ws;
    float* E1  = E0 + embElems;
    float* ACC = E1 + embElems;

    const int TPB = 256;
    int n4          = (int)(embElems / 4);                       // 1.6M
    int gridElem    = (int)((embElems * 16 / D_DIM + TPB - 1) / TPB); // N_ENT*16 thr
    int gridN4      = (n4 + TPB - 1) / TPB;
    long long sthreads = (long long)nnz * 16;
    int gridScatter = (int)((sthreads + TPB - 1) / TPB);

    // Build embeddings into E0 and ACC.
    cerp_build_kernel<<<gridElem, TPB, 0, stream>>>(Q, R, E0, ACC);

    // 3 LightGCN layers: dst = A * src (scatter-add), acc += dst, swap.
    float* src = E0;
    float* dst = E1;
    for (int l = 0; l < N_LAYERS; ++l) {
        zero_kernel<<<gridN4, TPB, 0, stream>>>((float4*)dst, n4);
        spmm_scatter_kernel<<<gridScatter, TPB, 0, stream>>>(src, dst, vals,
                                                             rows, cols, nnz);
        accum_kernel<<<gridN4, TPB, 0, stream>>>((float4*)ACC,
                                                 (const float4*)dst, n4);
        float* t = src; src = dst; dst = t;
    }

    // Batched per-row dots via WMMA: 16 rows per wave, 4 waves per block.
    int gridGamma = batch / 64;   // 16384/64 = 256
    gamma_wmma_kernel<<<gridGamma, 128, 0, stream>>>(ACC, users, items, out);
}